// EquivariantMessagePasser_16990890623300
// MI455X (gfx1250) — compile-verified
//
#include <hip/hip_runtime.h>
#include <math.h>

typedef __attribute__((ext_vector_type(2))) float v2f;
typedef __attribute__((ext_vector_type(8))) float v8f;

#define N_ATOMS 8000
#define N_EDGES 80000
#define HIDDEN  128

__device__ __forceinline__ float silu_f(float x) {
    // x * sigmoid(x). Use the hardware approximate reciprocal (v_rcp_f32,
    // a TRANS op that co-executes with the WMMA pipe) instead of IEEE
    // division, which expanded into a ~10-VALU div_scale/rcp/fma fixup
    // chain per element in the round-1 assembly.
    return x * __builtin_amdgcn_rcpf(1.0f + __expf(-x));
}

// ---------------------------------------------------------------------------
// Zero-init workspace region (pooled accumulators) — must run every call.
// ---------------------------------------------------------------------------
__global__ void zero_kernel(float* __restrict__ p, long long n) {
    long long i = (long long)blockIdx.x * blockDim.x + threadIdx.x;
    if (i < n) p[i] = 0.0f;
}

// ---------------------------------------------------------------------------
// Fused edge kernel:
//   H  = silu(rb @ W1)            (16x128 per wave, WMMA f32 16x16x4)
//   R  = H @ W2                   (16xK per wave, WMMA f32 16x16x4)
//   S  = sum_m Usub[i,j,m]*sh[m]  (d x d per edge)
//   pooled[center] += 0.25 * R[c] * (S @ nb)[i,k,c]   (f32 global atomics)
// Block = 128 threads = 4 waves, 16 edges per wave, 64 edges per block.
// ---------------------------------------------------------------------------
template<int NMAX, int K, int D, int MDIM, int MOFF>
__global__ __launch_bounds__(128) void edge_kernel(
    const float* __restrict__ rb, const float* __restrict__ sh,
    const float* __restrict__ feat, const float* __restrict__ W1,
    const float* __restrict__ W2, const float* __restrict__ U,
    const int* __restrict__ centers, const int* __restrict__ neighbors,
    float* __restrict__ pooled)
{
    constexpr int KT = (K + 15) / 16;        // N-tiles over K (zero-padded)
    __shared__ float Hlds[4][16][HIDDEN];
    __shared__ float Rlds[4][16][KT * 16];
    __shared__ float Slds[64][D * D];

    const int tid     = threadIdx.x;
    const int wave    = tid >> 5;
    const int lane    = tid & 31;
    const int halfrow = lane & 15;
    const bool hi     = lane >= 16;
    const int e0      = blockIdx.x * 64 + wave * 16;

    // ---- Stage 1: H = silu(rb @ W1) -------------------------------------
    for (int nt = 0; nt < HIDDEN / 16; ++nt) {
        v8f acc = {};
        for (int k0 = 0; k0 < NMAX; k0 += 4) {
            const int kk = hi ? k0 + 2 : k0;
            v2f a, b;
            a.x = (kk     < NMAX) ? rb[(e0 + halfrow) * NMAX + kk    ] : 0.0f;
            a.y = (kk + 1 < NMAX) ? rb[(e0 + halfrow) * NMAX + kk + 1] : 0.0f;
            const int n = nt * 16 + halfrow;
            b.x = (kk     < NMAX) ? W1[ kk      * HIDDEN + n] : 0.0f;
            b.y = (kk + 1 < NMAX) ? W1[(kk + 1) * HIDDEN + n] : 0.0f;
            acc = __builtin_amdgcn_wmma_f32_16x16x4_f32(
                false, a, false, b, (short)0, acc, false, false);
        }
#pragma unroll
        for (int r = 0; r < 8; ++r) {
            const int row = hi ? r + 8 : r;
            Hlds[wave][row][nt * 16 + halfrow] = silu_f(acc[r]);
        }
    }
    __syncthreads();

    // ---- Stage 2: R = H @ W2 --------------------------------------------
    for (int nt = 0; nt < KT; ++nt) {
        v8f acc = {};
        const int n = nt * 16 + halfrow;
        for (int k0 = 0; k0 < HIDDEN; k0 += 4) {
            const int kk = hi ? k0 + 2 : k0;
            v2f a, b;
            a.x = Hlds[wave][halfrow][kk];
            a.y = Hlds[wave][halfrow][kk + 1];
            b.x = (n < K) ? W2[ kk      * K + n] : 0.0f;
            b.y = (n < K) ? W2[(kk + 1) * K + n] : 0.0f;
            acc = __builtin_amdgcn_wmma_f32_16x16x4_f32(
                false, a, false, b, (short)0, acc, false, false);
        }
#pragma unroll
        for (int r = 0; r < 8; ++r) {
            const int row = hi ? r + 8 : r;
            Rlds[wave][row][nt * 16 + halfrow] = acc[r];
        }
    }

    // ---- S[i][j] = sum_m Usub[i,j,m] * sh[e,m] --------------------------
    if (tid < 64) {
        const int e = blockIdx.x * 64 + tid;
#pragma unroll
        for (int ij = 0; ij < D * D; ++ij) {
            float s = 0.0f;
#pragma unroll
            for (int m = 0; m < MDIM; ++m)
                s += U[ij * (D * D) + MOFF + m] * sh[e * MDIM + m];
            Slds[tid][ij] = s;
        }
    }
    __syncthreads();

    // ---- Scatter: comb = R[c] * (S @ nb); atomicAdd into pooled ---------
    for (int p = tid; p < 64 * K; p += 128) {
        const int el = p / K;
        const int c  = p - el * K;
        const int e  = blockIdx.x * 64 + el;
        const float scale = Rlds[el >> 4][el & 15][c] * 0.25f;  // MP_SCALING folded
        const float* nb = feat   + (size_t)neighbors[e] * (D * D * K) + c;
        float*       pb = pooled + (size_t)centers[e]   * (D * D * K) + c;
        float nbv[D][D];
#pragma unroll
        for (int j = 0; j < D; ++j)
#pragma unroll
            for (int k = 0; k < D; ++k)
                nbv[j][k] = nb[(j * D + k) * K];
#pragma unroll
        for (int i = 0; i < D; ++i) {
#pragma unroll
            for (int k = 0; k < D; ++k) {
                float a = 0.0f;
#pragma unroll
                for (int j = 0; j < D; ++j)
                    a += Slds[el][i * D + j] * nbv[j][k];
                atomicAdd(&pb[(i * D + k) * K], a * scale);
            }
        }
    }
}

// ---------------------------------------------------------------------------
// B1: c[n,m,k] = sum_{ij} U[i,j,m] * pooled[n,i,j,k]
// ---------------------------------------------------------------------------
template<int K, int D>
__global__ void u_contract_in(const float* __restrict__ pooled,
                              const float* __restrict__ U,
                              float* __restrict__ cbuf)
{
    const int idx = blockIdx.x * blockDim.x + threadIdx.x;
    if (idx >= N_ATOMS * K) return;
    const int n = idx / K;
    const int k = idx - n * K;
    const float* pp = pooled + (size_t)n * (D * D * K) + k;
    float pv[D * D];
#pragma unroll
    for (int ij = 0; ij < D * D; ++ij) pv[ij] = pp[ij * K];
    float* cp = cbuf + (size_t)n * (D * D * K) + k;
#pragma unroll
    for (int m = 0; m < D * D; ++m) {
        float s = 0.0f;
#pragma unroll
        for (int ij = 0; ij < D * D; ++ij)
            s += U[ij * (D * D) + m] * pv[ij];
        cp[m * K] = s;
    }
}

// ---------------------------------------------------------------------------
// B2: c2[n,m,q] = sum_k c[n,m,k] * Wlin[sl(m)][k,q]  — WMMA GEMM (M=atoms)
// grid = (125 atom-blocks, ceil(K/16) q-tiles, D*D m-values), 4 waves/block
// ---------------------------------------------------------------------------
template<int K, int D>
__global__ __launch_bounds__(128) void wlin_gemm(
    const float* __restrict__ cbuf, const float* __restrict__ Wlin,
    float* __restrict__ c2buf)
{
    const int tid     = threadIdx.x;
    const int wave    = tid >> 5;
    const int lane    = tid & 31;
    const int halfrow = lane & 15;
    const bool hi     = lane >= 16;
    const int m       = blockIdx.z;
    const int atom0   = (blockIdx.x * 4 + wave) * 16;

    int sl = 0;
    while ((sl + 1) * (sl + 1) <= m) ++sl;   // which Wlin block owns row m
    const float* W = Wlin + (size_t)sl * K * K;
    const int q = blockIdx.y * 16 + halfrow;

    const float* arow = cbuf + (size_t)(atom0 + halfrow) * (D * D * K) + m * K;

    v8f acc = {};
    for (int k0 = 0; k0 < K; k0 += 4) {       // K is a multiple of 4
        const int kk = hi ? k0 + 2 : k0;
        v2f a, b;
        a.x = arow[kk];
        a.y = arow[kk + 1];
        b.x = (q < K) ? W[ kk      * K + q] : 0.0f;
        b.y = (q < K) ? W[(kk + 1) * K + q] : 0.0f;
        acc = __builtin_amdgcn_wmma_f32_16x16x4_f32(
            false, a, false, b, (short)0, acc, false, false);
    }
    if (q < K) {
#pragma unroll
        for (int r = 0; r < 8; ++r) {
            const int row = hi ? r + 8 : r;
            c2buf[(size_t)(atom0 + row) * (D * D * K) + m * K + q] = acc[r];
        }
    }
}

// ---------------------------------------------------------------------------
// B3: out[n,i,j,k] = feat[n,i,j,k] + sum_m U[i,j,m] * c2[n,m,k]
// ---------------------------------------------------------------------------
template<int K, int D>
__global__ void u_contract_out(const float* __restrict__ c2buf,
                               const float* __restrict__ U,
                               const float* __restrict__ feat,
                               float* __restrict__ out)
{
    const int idx = blockIdx.x * blockDim.x + threadIdx.x;
    if (idx >= N_ATOMS * K) return;
    const int n = idx / K;
    const int k = idx - n * K;
    const float* cp = c2buf + (size_t)n * (D * D * K) + k;
    float cv[D * D];
#pragma unroll
    for (int m = 0; m < D * D; ++m) cv[m] = cp[m * K];
#pragma unroll
    for (int ij = 0; ij < D * D; ++ij) {
        float s = 0.0f;
#pragma unroll
        for (int m = 0; m < D * D; ++m)
            s += U[ij * (D * D) + m] * cv[m];
        const size_t off = (size_t)n * (D * D * K) + ij * K + k;
        out[off] = feat[off] + s;
    }
}

// ---------------------------------------------------------------------------
// Host-side orchestration
// ---------------------------------------------------------------------------
extern "C" void kernel_launch(void* const* d_in, const int* in_sizes, int n_in,
                              void* d_out, int out_size, void* d_ws, size_t ws_size,
                              hipStream_t stream) {
    (void)in_sizes; (void)n_in; (void)out_size; (void)ws_size;

    // Per-l geometry
    // l:      0     1     2     3
    // NMAX:  12    10     8     6
    // K:     96    48    24    16
    // D:      1     3     3     5
    // MDIM:   1     3     5     7   (2l+1)
    // MOFF:   0     1     4     9   (l*l)
    static const long long PSZ[4]  = { 8000LL *  1 * 96, 8000LL *  9 * 48,
                                       8000LL *  9 * 24, 8000LL * 25 * 16 };
    long long poff[4], total = 0;
    for (int l = 0; l < 4; ++l) { poff[l] = total; total += PSZ[l]; }  // total = 9,152,000

    float* pooled_base = (float*)d_ws;            // also aliased as c2 (consumed by then)
    float* cbuf_base   = (float*)d_ws + total;
    float* out_base    = (float*)d_out;

    const int* centers   = (const int*)d_in[27];
    const int* neighbors = (const int*)d_in[28];

    // Zero the pooled accumulators (required: harness does not re-zero ws)
    {
        const int threads = 256;
        const int blocks  = (int)((total + threads - 1) / threads);
        zero_kernel<<<blocks, threads, 0, stream>>>(pooled_base, total);
    }

#define RUN_L(L, NMAX, K, D, MDIM, MOFF, UIDX)                                          \
    do {                                                                                \
        const float* rb_   = (const float*)d_in[6 * (L) + 0];                           \
        const float* sh_   = (const float*)d_in[6 * (L) + 1];                           \
        const float* feat_ = (const float*)d_in[6 * (L) + 2];                           \
        const float* W1_   = (const float*)d_in[6 * (L) + 3];                           \
        const float* W2_   = (const float*)d_in[6 * (L) + 4];                           \
        const float* Wl_   = (const float*)d_in[6 * (L) + 5];                           \
        const float* U_    = (const float*)d_in[(UIDX)];                                \
        float* pooled_ = pooled_base + poff[(L)];                                       \
        float* cbuf_   = cbuf_base   + poff[(L)];                                       \
        float* c2_     = pooled_base + poff[(L)]; /* alias: pooled consumed by B1 */    \
        float* out_    = out_base    + poff[(L)];                                       \
        edge_kernel<NMAX, K, D, MDIM, MOFF><<<N_EDGES / 64, 128, 0, stream>>>(          \
            rb_, sh_, feat_, W1_, W2_, U_, centers, neighbors, pooled_);                \
        {                                                                               \
            const int nthr = N_ATOMS * (K);                                             \
            u_contract_in<K, D><<<(nthr + 255) / 256, 256, 0, stream>>>(                \
                pooled_, U_, cbuf_);                                                    \
            dim3 g2(N_ATOMS / 64, ((K) + 15) / 16, (D) * (D));                          \
            wlin_gemm<K, D><<<g2, 128, 0, stream>>>(cbuf_, Wl_, c2_);                   \
            u_contract_out<K, D><<<(nthr + 255) / 256, 256, 0, stream>>>(               \
                c2_, U_, feat_, out_);                                                  \
        }                                                                               \
    } while (0)

    RUN_L(0, 12, 96, 1, 1, 0, 24);   // U0
    RUN_L(1, 10, 48, 3, 3, 1, 25);   // U2
    RUN_L(2,  8, 24, 3, 5, 4, 25);   // U2
    RUN_L(3,  6, 16, 5, 7, 9, 26);   // U4

#undef RUN_L
}